// EdgeSAGELayer_47880295416397
// MI455X (gfx1250) — compile-verified
//
#include <hip/hip_runtime.h>

// EdgeSAGE layer for MI455X (gfx1250, wave32).
// Phase 1: weighted scatter-sum + degree count (atomics into L2-resident ws).
// Phase 2: fused mean + concat + GEMM(h @ W^T) + bias + ReLU + row-scale
//          using V_WMMA_F32_16X16X4_F32 (fp32-exact, memory-bound anyway).

#define N_NODES 100000
#define N_EDGES 1250000
#define E_FEAT  64
#define IN_CH   128
#define OUT_CH  128
#define K_TOT   (IN_CH + E_FEAT)   // 192

typedef __attribute__((ext_vector_type(2))) float v2f;
typedef __attribute__((ext_vector_type(8))) float v8f;

__global__ void edgesage_zero_kernel(float* __restrict__ ws, int n) {
    int i = blockIdx.x * blockDim.x + threadIdx.x;
    if (i < n) ws[i] = 0.0f;
}

// One wave32 per edge: lane L handles features [2L, 2L+1] (coalesced 8B load,
// two global f32 atomic adds into the 25.6MB seg_sum buffer, which is L2-resident).
__global__ void __launch_bounds__(256)
edgesage_scatter_kernel(const int*   __restrict__ edge_index,   // [2, E], row 0 = tgt
                        const float* __restrict__ edge_attr,    // [E, 64]
                        const float* __restrict__ edge_weight,  // [E]
                        float* __restrict__ seg_sum,            // [N, 64]
                        float* __restrict__ cnt) {              // [N]
    const int lane = threadIdx.x & 31;
    const int e    = (blockIdx.x * blockDim.x + threadIdx.x) >> 5;
    if (e >= N_EDGES) return;

    const int   tgt = edge_index[e];            // row 0 of [2, E]
    const float w   = edge_weight[e];

    const v2f ea = *(const v2f*)(edge_attr + (size_t)e * E_FEAT + lane * 2);
    float* dst = seg_sum + (size_t)tgt * E_FEAT + lane * 2;
    atomicAdd(dst + 0, ea.x * w);
    atomicAdd(dst + 1, ea.y * w);
    if (lane == 0) atomicAdd(cnt + tgt, 1.0f);
}

// One block (8 waves) per 16-node tile. Wave w computes output columns
// [16w, 16w+16). K = 192 in 48 steps of V_WMMA_F32_16X16X4_F32.
//
// A (16x4 f32) lane layout (ISA 7.12.2): lanes 0-15 -> M=lane, K=k0..k0+1 in
// VGPR0..1; lanes 16-31 -> M=lane-16, K=k0+2..k0+3.  => per-lane contiguous
// float2 at h[row][k0 + 2*(lane>>4)].
// B (4x16) mirrored: lane n holds column N=n&15, same K pairing => contiguous
// float2 of W[ncol][k0 + 2*(lane>>4)] since B[k][n] = W[n][k].
// C/D (16x16 f32): VGPR r -> M = r + 8*(lane>>4), N = lane&15.
__global__ void __launch_bounds__(256)
edgesage_gemm_kernel(const float* __restrict__ node_attr,   // [N, 128]
                     const float* __restrict__ seg_sum,     // [N, 64]
                     const float* __restrict__ cnt,         // [N]
                     const float* __restrict__ node_weight, // [N]
                     const float* __restrict__ W,           // [128, 192]
                     const float* __restrict__ bias,        // [128]
                     float* __restrict__ out) {             // [N, 128]
    const int tile = blockIdx.x;          // 6250 tiles of 16 nodes
    const int lane = threadIdx.x & 31;
    const int wv   = threadIdx.x >> 5;    // 0..7 -> output column group
    const int half = lane >> 4;           // 0 or 1
    const int l16  = lane & 15;

    const int row  = l16;                 // A-matrix row owned by this lane
    const int node = tile * 16 + row;     // exact: 100000 % 16 == 0
    const int ncol = wv * 16 + l16;       // B/C/D column owned by this lane

    const float c    = cnt[node];
    const float invc = 1.0f / fmaxf(c, 1.0f);   // scatter-mean; zero-degree -> 0/1

    const float* aptr = node_attr + (size_t)node * IN_CH;
    const float* sptr = seg_sum   + (size_t)node * E_FEAT;
    const float* wptr = W         + (size_t)ncol * K_TOT;

    const int koff = half * 2;            // lane's K sub-offset within each k0 step
    v8f acc = {0.f, 0.f, 0.f, 0.f, 0.f, 0.f, 0.f, 0.f};

    // K = 0..127: A from node_attr
#pragma unroll
    for (int k0 = 0; k0 < IN_CH; k0 += 4) {
        v2f a  = *(const v2f*)(aptr + k0 + koff);
        v2f bb = *(const v2f*)(wptr + k0 + koff);
        acc = __builtin_amdgcn_wmma_f32_16x16x4_f32(
            /*neg_a=*/false, a, /*neg_b=*/false, bb,
            /*c_mod=*/(short)0, acc, /*reuse_a=*/false, /*reuse_b=*/false);
    }
    // K = 128..191: A = seg_sum * invc (the scatter-mean, fused)
#pragma unroll
    for (int k0 = 0; k0 < E_FEAT; k0 += 4) {
        v2f s  = *(const v2f*)(sptr + k0 + koff);
        v2f a; a.x = s.x * invc; a.y = s.y * invc;
        v2f bb = *(const v2f*)(wptr + IN_CH + k0 + koff);
        acc = __builtin_amdgcn_wmma_f32_16x16x4_f32(
            false, a, false, bb, (short)0, acc, false, false);
    }

    // Epilogue: bias + ReLU + node_weight scale, per C/D lane layout.
    const float bval = bias[ncol];
#pragma unroll
    for (int r = 0; r < 8; ++r) {
        const int m     = r + 8 * half;
        const int onode = tile * 16 + m;
        float v = acc[r] + bval;
        v = v > 0.0f ? v : 0.0f;
        v *= node_weight[onode];
        out[(size_t)onode * OUT_CH + ncol] = v;
    }
}

extern "C" void kernel_launch(void* const* d_in, const int* in_sizes, int n_in,
                              void* d_out, int out_size, void* d_ws, size_t ws_size,
                              hipStream_t stream) {
    const int*   edge_index  = (const int*)  d_in[0]; // [2, E] int32
    const float* edge_attr   = (const float*)d_in[1]; // [E, 64]
    const float* node_attr   = (const float*)d_in[2]; // [N, 128]
    const float* edge_weight = (const float*)d_in[3]; // [E]
    const float* node_weight = (const float*)d_in[4]; // [N]
    const float* W           = (const float*)d_in[5]; // [128, 192]
    const float* bias        = (const float*)d_in[6]; // [128]
    float*       out         = (float*)d_out;         // [N, 128]

    float* seg_sum = (float*)d_ws;                           // N*64 floats
    float* cnt     = seg_sum + (size_t)N_NODES * E_FEAT;     // N floats

    // 1) zero accumulators (ws is poisoned; must re-zero every call)
    const int nz = N_NODES * E_FEAT + N_NODES;
    edgesage_zero_kernel<<<(nz + 255) / 256, 256, 0, stream>>>(seg_sum, nz);

    // 2) scatter: one wave per edge -> E*32 threads
    const long long nthreads = (long long)N_EDGES * 32;
    const int nblocks = (int)((nthreads + 255) / 256);       // 156250
    edgesage_scatter_kernel<<<nblocks, 256, 0, stream>>>(
        edge_index, edge_attr, edge_weight, seg_sum, cnt);

    // 3) fused mean + concat + GEMM + bias + ReLU + scale
    edgesage_gemm_kernel<<<N_NODES / 16, 256, 0, stream>>>(
        node_attr, seg_sum, cnt, node_weight, W, bias, out);
}